// S4DKernel_30236569763907
// MI455X (gfx1250) — compile-verified
//
#include <hip/hip_runtime.h>

// S4D kernel generation for MI455X (gfx1250, wave32).
//
// K[h,l] = Re( sum_n Cc[h,n] * w[h,n]^l ),  w = exp(dt_h * (A_re + i*A_im))
//
// Factor l = 256*t + 16*mi + ni and compute each 16x16 output tile as an
// fp32 WMMA chain:  D = A(16x128) x B(128x16), K-dim = 64 states x {re,im}.
// Uses V_WMMA_F32_16X16X4_F32 (32 chained steps per tile) -> full f32
// precision, only 64 complex exps per channel, everything else is cheap
// complex-mul recurrences. One wave per channel; 1024 single-wave blocks.

typedef float v2f __attribute__((ext_vector_type(2)));
typedef float v8f __attribute__((ext_vector_type(8)));

#define HDIM    1024
#define NSTATE  64
#define LLEN    2048
#define PITCH   17      // LDS row pitch (floats) to spread banks

__global__ __launch_bounds__(32) void s4d_wmma_kernel(
    const float* __restrict__ log_dt,
    const float* __restrict__ A_re,
    const float* __restrict__ A_im,
    const float* __restrict__ Cmat,
    float* __restrict__ out)
{
    // A^T (k-major) and B, both 128 x 16 f32, padded.
    __shared__ float AT[128 * PITCH];
    __shared__ float BT[128 * PITCH];

    const int h    = blockIdx.x;
    const int lane = threadIdx.x;          // 0..31 (wave32)

    const float dt = __expf(log_dt[h]);

    // Each lane owns states n = lane and n = lane + 32.
    float w_re[2], w_im[2];        // w = exp(dt*A)
    float w16_re[2], w16_im[2];    // w^16
    float W256_re[2], W256_im[2];  // w^256
    float base_re[2], base_im[2];  // Cc * w^(256*t), advanced per tile

    #pragma unroll
    for (int j = 0; j < 2; ++j) {
        const int n  = lane + 32 * j;
        const float ar = A_re[h * NSTATE + n];
        const float ai = A_im[h * NSTATE + n];
        const float r  = __expf(dt * ar);
        const float th = dt * ai;
        w_re[j] = r * __cosf(th);
        w_im[j] = r * __sinf(th);
        base_re[j] = Cmat[(h * NSTATE + n) * 2 + 0];
        base_im[j] = Cmat[(h * NSTATE + n) * 2 + 1];
    }

    // Build tile-invariant B: BT[2n][ni] = Re(w^ni), BT[2n+1][ni] = -Im(w^ni).
    #pragma unroll
    for (int j = 0; j < 2; ++j) {
        const int n = lane + 32 * j;
        float cr = 1.0f, ci = 0.0f;        // w^0
        #pragma unroll
        for (int ni = 0; ni < 16; ++ni) {
            BT[(2 * n    ) * PITCH + ni] =  cr;
            BT[(2 * n + 1) * PITCH + ni] = -ci;
            const float nr = cr * w_re[j] - ci * w_im[j];
            const float ni_ = cr * w_im[j] + ci * w_re[j];
            cr = nr; ci = ni_;
        }
        w16_re[j] = cr; w16_im[j] = ci;    // now w^16
        // W256 = (w^16)^16 via 4 squarings
        float sr = cr, si = ci;
        #pragma unroll
        for (int s = 0; s < 4; ++s) {
            const float tr = sr * sr - si * si;
            const float ti = 2.0f * sr * si;
            sr = tr; si = ti;
        }
        W256_re[j] = sr; W256_im[j] = si;
    }

    const int hf  = lane >> 4;             // half-wave select (0/1)
    const int m16 = lane & 15;             // row/col within half

    for (int t = 0; t < 8; ++t) {
        // Fill A^T for this tile: AT[2n][mi] = Re(base * (w^16)^mi), AT[2n+1][mi] = Im.
        #pragma unroll
        for (int j = 0; j < 2; ++j) {
            const int n = lane + 32 * j;
            float pr = base_re[j], pi = base_im[j];
            #pragma unroll
            for (int mi = 0; mi < 16; ++mi) {
                AT[(2 * n    ) * PITCH + mi] = pr;
                AT[(2 * n + 1) * PITCH + mi] = pi;
                const float nr = pr * w16_re[j] - pi * w16_im[j];
                const float ni_ = pr * w16_im[j] + pi * w16_re[j];
                pr = nr; pi = ni_;
            }
            // advance base to next tile: base *= w^256
            const float nr = base_re[j] * W256_re[j] - base_im[j] * W256_im[j];
            const float ni_ = base_re[j] * W256_im[j] + base_im[j] * W256_re[j];
            base_re[j] = nr; base_im[j] = ni_;
        }
        __syncthreads();

        // 32-step WMMA chain over K = 128 (f32, exact).
        v8f acc = {0.f, 0.f, 0.f, 0.f, 0.f, 0.f, 0.f, 0.f};
        #pragma unroll 4
        for (int k0 = 0; k0 < 128; k0 += 4) {
            // A operand layout (16x4 f32): VGPR0 = K{k0 | k0+2}, VGPR1 = K{k0+1 | k0+3}
            v2f a, b;
            a.x = AT[(k0 + 2 * hf    ) * PITCH + m16];
            a.y = AT[(k0 + 2 * hf + 1) * PITCH + m16];
            b.x = BT[(k0 + 2 * hf    ) * PITCH + m16];
            b.y = BT[(k0 + 2 * hf + 1) * PITCH + m16];
            acc = __builtin_amdgcn_wmma_f32_16x16x4_f32(
                /*neg_a=*/false, a, /*neg_b=*/false, b,
                /*c_mod=*/(short)0, acc, /*reuse_a=*/false, /*reuse_b=*/false);
        }

        // Store: D VGPR r -> M = r + 8*hf, N = lane&15;  l = 256*t + 16*M + N.
        float* op = out + h * LLEN + t * 256;
        #pragma unroll
        for (int r = 0; r < 8; ++r) {
            op[(r + 8 * hf) * 16 + m16] = acc[r];
        }
        __syncthreads();   // AT rewritten next tile
    }
}

extern "C" void kernel_launch(void* const* d_in, const int* in_sizes, int n_in,
                              void* d_out, int out_size, void* d_ws, size_t ws_size,
                              hipStream_t stream) {
    (void)in_sizes; (void)n_in; (void)out_size; (void)d_ws; (void)ws_size;
    const float* log_dt = (const float*)d_in[0];
    const float* A_re   = (const float*)d_in[1];
    const float* A_im   = (const float*)d_in[2];
    const float* Cmat   = (const float*)d_in[3];
    // d_in[4] is L (==2048), compile-time constant here.
    float* out = (float*)d_out;

    s4d_wmma_kernel<<<dim3(HDIM), dim3(32), 0, stream>>>(
        log_dt, A_re, A_im, Cmat, out);
}